// CascadingSystem_70703751627396
// MI455X (gfx1250) — compile-verified
//
#include <hip/hip_runtime.h>

typedef __attribute__((ext_vector_type(2))) float v2f;
typedef __attribute__((ext_vector_type(8))) float v8f;

#define D_TOTAL      150528          // 3*224*224
#define B_ROWS       256
#define K_PER_BLOCK  1024
#define K_PER_WAVE   256             // 4 waves per block
#define N_KCHUNK     (D_TOTAL / K_PER_BLOCK)   // 147
#define N_SLAB       (B_ROWS / 16)             // 16

// ---------------------------------------------------------------------------
// Zero the [256 x 4] fp32 partial-logit accumulator in workspace.
// ---------------------------------------------------------------------------
__global__ void zero_acc_kernel(float* __restrict__ acc) {
    int i = blockIdx.x * blockDim.x + threadIdx.x;
    if (i < B_ROWS * 4) acc[i] = 0.0f;
}

// ---------------------------------------------------------------------------
// Main bandwidth kernel: [256 x 150528] @ [150528 x 4] using
// V_WMMA_F32_16X16X4_F32 (fp32 in / fp32 accumulate, exact numerics).
//
// grid  = (16 row-slabs, 147 K-chunks), block = 128 threads (4 waves).
// Each wave owns 16 batch rows (M) and a 256-element K range (64 WMMA steps).
//
// A-tile per lane: x[row = lane&15, k + 2*(lane>>4) + {0,1}]  (b64 load)
// B-tile per lane: col n = lane&15.
//   n==0,1 -> W1 column n ; n==2,3 -> W2 column n-2.
//   n>=4   -> W2 column (n&1): harmless in-bounds garbage, because
//             accumulator columns 4..15 are NEVER read by the reduction.
//   => no EXEC masking / branches anywhere in the inner loop.
//
// Wave partials (cols 0..3 only) reduced via LDS ds_add_f32, then one
// global_atomic_add_f32 per (row, col) per workgroup.
// ---------------------------------------------------------------------------
__global__ __launch_bounds__(128)
void cascade_gemm_wmma_kernel(const float* __restrict__ x,
                              const float* __restrict__ W1,
                              const float* __restrict__ W2,
                              float* __restrict__ acc) {
    __shared__ float lds_acc[16 * 4];

    const int tid   = threadIdx.x;
    const int lane  = tid & 31;
    const int wave  = tid >> 5;           // 0..3
    const int slab  = blockIdx.x;         // rows slab*16 .. slab*16+15
    const int kbase = blockIdx.y * K_PER_BLOCK + wave * K_PER_WAVE;

    if (tid < 64) lds_acc[tid] = 0.0f;
    __syncthreads();

    // ---- A operand addressing (16x4 fp32 tile, ISA layout) ----
    const int m     = lane & 15;                 // row within slab
    const int khalf = (lane >> 4) * 2;           // 0 or 2
    const float* __restrict__ xptr =
        x + (long)(slab * 16 + m) * (long)D_TOTAL + (long)(kbase + khalf);

    // ---- B operand addressing (4x16 fp32 tile): column n = lane&15 ----
    // Branch-free column select; n>=4 reads valid-but-unused W2 data.
    const int n = lane & 15;
    const float* __restrict__ wptr =
        ((n < 2) ? (W1 + n) : (W2 + (n & 1))) + (long)(kbase + khalf) * 2;
    // wptr[0] = W[k+khalf, col], wptr[2] = W[k+khalf+1, col]

    v8f c = {};   // 16x16 fp32 accumulator (8 VGPRs)

    #pragma unroll 8
    for (int kk = 0; kk < K_PER_WAVE; kk += 4) {
        // A: contiguous 8B per lane -> 256B of x per WMMA across the wave
        v2f a = *(const v2f*)xptr;

        // B: two dword loads from the (L2-resident) weight columns
        v2f b;
        b.x = wptr[0];
        b.y = wptr[2];

        // D = A(16x4) x B(4x16) + C   -> v_wmma_f32_16x16x4_f32
        c = __builtin_amdgcn_wmma_f32_16x16x4_f32(
                /*neg_a=*/false, a, /*neg_b=*/false, b,
                /*c_mod=*/(short)0, c, /*reuse_a=*/false, /*reuse_b=*/false);

        xptr += 4;       // advance 4 K (folds into load offsets under unroll)
        wptr += 8;       // 4 K * 2 cols
    }

    // C/D layout: lane holds column N = lane&15; VGPR i -> row M=(lane>>4)*8+i
    if (n < 4) {
        const int mbase = (lane >> 4) * 8;
        #pragma unroll
        for (int i = 0; i < 8; ++i) {
            atomicAdd(&lds_acc[(mbase + i) * 4 + n], c[i]);   // ds_add_f32
        }
    }
    __syncthreads();

    if (tid < 64) {
        const int mr = tid >> 2;
        const int nc = tid & 3;
        atomicAdd(&acc[(slab * 16 + mr) * 4 + nc], lds_acc[tid]); // global atomic
    }
}

// ---------------------------------------------------------------------------
// Epilogue: biases, softmax confidence, cascade blend. 256 rows.
// ---------------------------------------------------------------------------
__global__ void cascade_epilogue_kernel(const float* __restrict__ acc,
                                        const float* __restrict__ b1,
                                        const float* __restrict__ b2,
                                        float* __restrict__ out) {
    int r = blockIdx.x * blockDim.x + threadIdx.x;
    if (r >= B_ROWS) return;

    float t0 = acc[r * 4 + 0] + b1[0];
    float t1 = acc[r * 4 + 1] + b1[1];
    float f0 = acc[r * 4 + 2] + b2[0];
    float f1 = acc[r * 4 + 3] + b2[1];

    float mx = fmaxf(t0, t1);
    float e0 = __expf(t0 - mx);
    float e1 = __expf(t1 - mx);
    float conf = fmaxf(e0, e1) / (e0 + e1);

    float o0, o1;
    if (conf > 0.95f) {
        o0 = t0; o1 = t1;
    } else {
        o0 = 0.7f * t0 + 0.3f * f0;
        o1 = 0.7f * t1 + 0.3f * f1;
    }
    out[r * 2 + 0] = o0;
    out[r * 2 + 1] = o1;
}

// ---------------------------------------------------------------------------
extern "C" void kernel_launch(void* const* d_in, const int* in_sizes, int n_in,
                              void* d_out, int out_size, void* d_ws, size_t ws_size,
                              hipStream_t stream) {
    const float* x  = (const float*)d_in[0];   // [256,3,224,224]
    const float* W1 = (const float*)d_in[1];   // [150528,2]
    const float* b1 = (const float*)d_in[2];   // [2]
    const float* W2 = (const float*)d_in[3];   // [150528,2]
    const float* b2 = (const float*)d_in[4];   // [2]
    float* out = (float*)d_out;                // [256,2]
    float* acc = (float*)d_ws;                 // [256,4] fp32 partial logits

    zero_acc_kernel<<<(B_ROWS * 4 + 255) / 256, 256, 0, stream>>>(acc);

    dim3 grid(N_SLAB, N_KCHUNK);               // 16 x 147 workgroups
    cascade_gemm_wmma_kernel<<<grid, 128, 0, stream>>>(x, W1, W2, acc);

    cascade_epilogue_kernel<<<(B_ROWS + 127) / 128, 128, 0, stream>>>(acc, b1, b2, out);
}